// GraphDecoder_61306363183554
// MI455X (gfx1250) — compile-verified
//
#include <hip/hip_runtime.h>
#include <hip/hip_bf16.h>
#include <stdint.h>

#define LATN 128
#define MATN 8
#define KTOT 136   // LATN + MATN
#define OUTN 3
#define LDSPAD 137 // pad row stride to avoid LDS bank conflicts

typedef __attribute__((ext_vector_type(2))) float v2f;
typedef __attribute__((ext_vector_type(8))) float v8f;

// ---------------------------------------------------------------------------
// 1) degree / normalization
// ---------------------------------------------------------------------------
__global__ __launch_bounds__(256) void init_deg_kernel(float* deg, int n) {
    int i = blockIdx.x * blockDim.x + threadIdx.x;
    if (i < n) deg[i] = 1.0f;  // self-loop contributes 1 to every node's degree
}

__global__ __launch_bounds__(256) void count_deg_kernel(const int* __restrict__ dst,
                                                        float* __restrict__ deg, int ne) {
    int t = blockIdx.x * blockDim.x + threadIdx.x;
    int stride = gridDim.x * blockDim.x;
    for (int e = t; e < ne; e += stride) {
        atomicAdd(&deg[dst[e]], 1.0f);
    }
}

__global__ __launch_bounds__(256) void dinv_kernel(float* deg, int n) {
    int i = blockIdx.x * blockDim.x + threadIdx.x;
    if (i < n) {
        float d = deg[i];
        deg[i] = (d > 0.0f) ? rsqrtf(d) : 0.0f;  // in-place: deg -> dinv
    }
}

// ---------------------------------------------------------------------------
// 2) h = concat(z, mat) @ W1  via V_WMMA_F32_16X16X4_F32
//    block = 256 threads = 8 waves; block owns 16 rows (nodes),
//    wave w owns output columns [16w, 16w+16). K loop: 136 in steps of 4.
// ---------------------------------------------------------------------------
__global__ __launch_bounds__(256) void gemm1_wmma_kernel(const float* __restrict__ z,
                                                         const float* __restrict__ mat,
                                                         const float* __restrict__ W1,
                                                         float* __restrict__ h, int n) {
    __shared__ float sA[16 * LDSPAD];  // 16 x 136 A-tile (padded stride)

    const int rowTile = blockIdx.x * 16;
    const int tid = threadIdx.x;

    // cooperative stage of A-tile = concat(z[16,128], mat[16,8])
    for (int i = tid; i < 16 * LATN; i += 256) {
        int r = i >> 7, c = i & (LATN - 1);
        int row = rowTile + r;
        sA[r * LDSPAD + c] = (row < n) ? z[(size_t)row * LATN + c] : 0.0f;
    }
    for (int i = tid; i < 16 * MATN; i += 256) {
        int r = i >> 3, c = i & (MATN - 1);
        int row = rowTile + r;
        sA[r * LDSPAD + LATN + c] = (row < n) ? mat[(size_t)row * MATN + c] : 0.0f;
    }
    __syncthreads();

    const int wave = tid >> 5;          // 0..7 -> column tile
    const int lane = tid & 31;
    const int colBase = wave * 16;
    const int m  = lane & 15;           // row (A) / col (B,C,D) index within tile
    const int kh = (lane >> 4) * 2;     // lanes 0-15: K={0,1}; lanes 16-31: K={2,3}

    v8f acc = {};
    for (int k = 0; k < KTOT; k += 4) {
        v2f a, b;
        // A 16x4 fp32 layout: VGPR0 = K=kh, VGPR1 = K=kh+1 for row m
        a.x = sA[m * LDSPAD + k + kh];
        a.y = sA[m * LDSPAD + k + kh + 1];
        // B 4x16 fp32 layout: VGPR0 = B[kh][colBase+m], VGPR1 = B[kh+1][colBase+m]
        b.x = W1[(size_t)(k + kh) * LATN + colBase + m];
        b.y = W1[(size_t)(k + kh + 1) * LATN + colBase + m];
        acc = __builtin_amdgcn_wmma_f32_16x16x4_f32(false, a, false, b,
                                                    (short)0, acc, false, false);
    }

    // C/D layout: VGPR v holds row (v + 8*(lane>>4)), col (colBase + (lane&15))
    const int rowOff = (lane >> 4) * 8;
    const int col = colBase + m;
#pragma unroll
    for (int v = 0; v < 8; ++v) {
        int row = rowTile + rowOff + v;
        if (row < n) h[(size_t)row * LATN + col] = acc[v];
    }
}

// ---------------------------------------------------------------------------
// 3) layer-1 aggregation: agg1 = b1 + dinv^2 * h  (self loop)  + edge scatter
// ---------------------------------------------------------------------------
__global__ __launch_bounds__(256) void init_agg1_kernel(const float* __restrict__ h,
                                                        const float* __restrict__ dinv,
                                                        const float* __restrict__ b1,
                                                        float* __restrict__ agg, int n) {
    int idx = blockIdx.x * blockDim.x + threadIdx.x;
    if (idx < n * LATN) {
        int i = idx >> 7, f = idx & (LATN - 1);
        float w = dinv[i] * dinv[i];
        agg[idx] = b1[f] + w * h[idx];
    }
}

// one wave per edge: 32 lanes x float4 = full 512B feature row, coalesced
__global__ __launch_bounds__(256) void scatter1_kernel(const float* __restrict__ h,
                                                       const float* __restrict__ dinv,
                                                       const int* __restrict__ src,
                                                       const int* __restrict__ dst,
                                                       float* __restrict__ agg, int ne) {
    int gtid = blockIdx.x * blockDim.x + threadIdx.x;
    int wid = gtid >> 5;
    int lane = threadIdx.x & 31;
    int nWaves = (gridDim.x * blockDim.x) >> 5;

    for (int e = wid; e < ne; e += nWaves) {
        int s = src[e], d = dst[e];
        // prefetch next edge's source row into cache (global_prefetch_b8)
        int en = e + nWaves;
        if (en < ne) {
            int sn = src[en];
            __builtin_prefetch(h + (size_t)sn * LATN + lane * 4, 0, 0);
        }
        float w = dinv[s] * dinv[d];
        const float4* hs = (const float4*)(h + (size_t)s * LATN);
        float4 v = hs[lane];
        float* ad = agg + (size_t)d * LATN + lane * 4;
        atomicAdd(ad + 0, w * v.x);
        atomicAdd(ad + 1, w * v.y);
        atomicAdd(ad + 2, w * v.z);
        atomicAdd(ad + 3, w * v.w);
    }
}

// ---------------------------------------------------------------------------
// 4) h2 = relu(agg1) @ W2  (128 -> 3), wave per node, shuffle reduction
// ---------------------------------------------------------------------------
__global__ __launch_bounds__(256) void gemm2_kernel(const float* __restrict__ agg,
                                                    const float* __restrict__ W2,
                                                    float* __restrict__ h2, int n) {
    int gtid = blockIdx.x * blockDim.x + threadIdx.x;
    int wid = gtid >> 5;
    int lane = threadIdx.x & 31;
    int nWaves = (gridDim.x * blockDim.x) >> 5;

    for (int i = wid; i < n; i += nWaves) {
        const float4* xr = (const float4*)(agg + (size_t)i * LATN);
        float4 v = xr[lane];
        v.x = fmaxf(v.x, 0.0f);
        v.y = fmaxf(v.y, 0.0f);
        v.z = fmaxf(v.z, 0.0f);
        v.w = fmaxf(v.w, 0.0f);
        int c = lane * 4;  // K offset
        float s0 = 0.f, s1 = 0.f, s2 = 0.f;
        // W2 row-major [128,3]
        s0 = fmaf(v.x, W2[(c + 0) * OUTN + 0], s0);
        s1 = fmaf(v.x, W2[(c + 0) * OUTN + 1], s1);
        s2 = fmaf(v.x, W2[(c + 0) * OUTN + 2], s2);
        s0 = fmaf(v.y, W2[(c + 1) * OUTN + 0], s0);
        s1 = fmaf(v.y, W2[(c + 1) * OUTN + 1], s1);
        s2 = fmaf(v.y, W2[(c + 1) * OUTN + 2], s2);
        s0 = fmaf(v.z, W2[(c + 2) * OUTN + 0], s0);
        s1 = fmaf(v.z, W2[(c + 2) * OUTN + 1], s1);
        s2 = fmaf(v.z, W2[(c + 2) * OUTN + 2], s2);
        s0 = fmaf(v.w, W2[(c + 3) * OUTN + 0], s0);
        s1 = fmaf(v.w, W2[(c + 3) * OUTN + 1], s1);
        s2 = fmaf(v.w, W2[(c + 3) * OUTN + 2], s2);
#pragma unroll
        for (int off = 16; off > 0; off >>= 1) {
            s0 += __shfl_down(s0, off, 32);
            s1 += __shfl_down(s1, off, 32);
            s2 += __shfl_down(s2, off, 32);
        }
        if (lane == 0) {
            h2[(size_t)i * OUTN + 0] = s0;
            h2[(size_t)i * OUTN + 1] = s1;
            h2[(size_t)i * OUTN + 2] = s2;
        }
    }
}

// ---------------------------------------------------------------------------
// 5) layer-2 aggregation: out = b2 + dinv^2 * h2 + edge scatter
// ---------------------------------------------------------------------------
__global__ __launch_bounds__(256) void init_out_kernel(const float* __restrict__ h2,
                                                       const float* __restrict__ dinv,
                                                       const float* __restrict__ b2,
                                                       float* __restrict__ out, int n) {
    int i = blockIdx.x * blockDim.x + threadIdx.x;
    if (i < n) {
        float w = dinv[i] * dinv[i];
        out[(size_t)i * OUTN + 0] = b2[0] + w * h2[(size_t)i * OUTN + 0];
        out[(size_t)i * OUTN + 1] = b2[1] + w * h2[(size_t)i * OUTN + 1];
        out[(size_t)i * OUTN + 2] = b2[2] + w * h2[(size_t)i * OUTN + 2];
    }
}

__global__ __launch_bounds__(256) void scatter2_kernel(const float* __restrict__ h2,
                                                       const float* __restrict__ dinv,
                                                       const int* __restrict__ src,
                                                       const int* __restrict__ dst,
                                                       float* __restrict__ out, int ne) {
    int t = blockIdx.x * blockDim.x + threadIdx.x;
    int stride = gridDim.x * blockDim.x;
    for (int e = t; e < ne; e += stride) {
        int s = src[e], d = dst[e];
        float w = dinv[s] * dinv[d];
        atomicAdd(out + (size_t)d * OUTN + 0, w * h2[(size_t)s * OUTN + 0]);
        atomicAdd(out + (size_t)d * OUTN + 1, w * h2[(size_t)s * OUTN + 1]);
        atomicAdd(out + (size_t)d * OUTN + 2, w * h2[(size_t)s * OUTN + 2]);
    }
}

// ---------------------------------------------------------------------------
// host launch
// ---------------------------------------------------------------------------
static inline size_t alignUp(size_t x) { return (x + 255) & ~size_t(255); }

extern "C" void kernel_launch(void* const* d_in, const int* in_sizes, int n_in,
                              void* d_out, int out_size, void* d_ws, size_t ws_size,
                              hipStream_t stream) {
    const float* z   = (const float*)d_in[0];
    const int*   ei  = (const int*)d_in[1];
    const float* mp  = (const float*)d_in[2];
    const float* W1  = (const float*)d_in[3];
    const float* b1  = (const float*)d_in[4];
    const float* W2  = (const float*)d_in[5];
    const float* b2  = (const float*)d_in[6];
    float* out = (float*)d_out;

    const int n  = in_sizes[0] / LATN;   // nodes
    const int ne = in_sizes[1] / 2;      // directed edges
    const int* src = ei;                 // edge_index[0, :]
    const int* dst = ei + ne;            // edge_index[1, :]

    // workspace carve-up: dinv[n], h[n*128], agg1[n*128], h2[n*3]
    char* ws = (char*)d_ws;
    size_t off = 0;
    float* dinv = (float*)(ws + off); off += alignUp((size_t)n * sizeof(float));
    float* h    = (float*)(ws + off); off += alignUp((size_t)n * LATN * sizeof(float));
    float* agg1 = (float*)(ws + off); off += alignUp((size_t)n * LATN * sizeof(float));
    float* h2   = (float*)(ws + off); off += alignUp((size_t)n * OUTN * sizeof(float));
    (void)ws_size; (void)n_in; (void)out_size;

    const int tb = 256;

    // 1) symmetric GCN normalization
    init_deg_kernel<<<(n + tb - 1) / tb, tb, 0, stream>>>(dinv, n);
    count_deg_kernel<<<2048, tb, 0, stream>>>(dst, dinv, ne);
    dinv_kernel<<<(n + tb - 1) / tb, tb, 0, stream>>>(dinv, n);

    // 2) dense layer 1 on WMMA
    gemm1_wmma_kernel<<<(n + 15) / 16, tb, 0, stream>>>(z, mp, W1, h, n);

    // 3) layer-1 aggregation (self loop init + atomic edge scatter)
    init_agg1_kernel<<<((size_t)n * LATN + tb - 1) / tb, tb, 0, stream>>>(h, dinv, b1, agg1, n);
    scatter1_kernel<<<4096, tb, 0, stream>>>(h, dinv, src, dst, agg1, ne);

    // 4) dense layer 2 (relu fused)
    gemm2_kernel<<<2048, tb, 0, stream>>>(agg1, W2, h2, n);

    // 5) layer-2 aggregation directly into d_out
    init_out_kernel<<<(n + tb - 1) / tb, tb, 0, stream>>>(h2, dinv, b2, out, n);
    scatter2_kernel<<<2048, tb, 0, stream>>>(h2, dinv, src, dst, out, ne);
}